// GTLayer_8486855377181
// MI455X (gfx1250) — compile-verified
//
#include <hip/hip_runtime.h>
#include <hip/hip_bf16.h>
#include <math.h>

#define N_NODES 100000
#define N_EDGES 800000
#define D_MODEL 128
#define N_HEAD  8
#define D_HEAD  16
#define LN_EPS  1e-6f
#define LDS_STRIDE 132   // 128 + 4 pad: row stride mod 64 banks = 4 -> conflict-free fragment reads

typedef float v2f __attribute__((ext_vector_type(2)));
typedef float v8f __attribute__((ext_vector_type(8)));

// ---------------------------------------------------------------------------
// Kernel 1: Q = E@qW, K = E@kW, V = E@vW via V_WMMA_F32_16X16X4_F32 (wave32).
// Block = 256 threads = 8 waves; block owns one 16-row M-tile (100000 = 6250*16
// exactly, so no guards and EXEC is all-ones for WMMA). Wave w owns N-tile
// n0 = 16*w. Also initializes res = embeds and norm = 0 for later phases.
// ---------------------------------------------------------------------------
__global__ __launch_bounds__(256) void gt_qkv_gemm(
    const float* __restrict__ E,  const float* __restrict__ qW,
    const float* __restrict__ kW, const float* __restrict__ vW,
    float* __restrict__ Q, float* __restrict__ K, float* __restrict__ V,
    float* __restrict__ res, float* __restrict__ norm)
{
    __shared__ float ldsA[16 * LDS_STRIDE];
    const int m0  = blockIdx.x * 16;
    const int tid = threadIdx.x;

    // Stage 16x128 f32 tile (512 float4) into LDS; fused res-init (residual).
    #pragma unroll
    for (int i = 0; i < 2; ++i) {
        int idx = tid + i * 256;          // float4 index 0..511
        int r   = idx >> 5;               // 32 float4 per row
        int c4  = idx & 31;
        float4 val = ((const float4*)(E + (size_t)(m0 + r) * D_MODEL))[c4];
        ((float4*)(res + (size_t)(m0 + r) * D_MODEL))[c4] = val;
        float* dst = &ldsA[r * LDS_STRIDE + c4 * 4];
        dst[0] = val.x; dst[1] = val.y; dst[2] = val.z; dst[3] = val.w;
    }
    // Zero softmax-denominator accumulator: 16 rows * 8 heads = 128 per block.
    if (tid < 16 * N_HEAD) norm[(size_t)blockIdx.x * 128 + tid] = 0.0f;
    __syncthreads();

    const int lane = tid & 31;
    const int wave = tid >> 5;
    const int n0   = wave * 16;
    const int half = lane >> 4;      // 0 -> K pair {0,1}, 1 -> K pair {2,3}
    const int l15  = lane & 15;

    v8f accq = {}; v8f acck = {}; v8f accv = {};
    const float* arow = &ldsA[l15 * LDS_STRIDE + 2 * half];

    #pragma unroll 4
    for (int k0 = 0; k0 < D_MODEL; k0 += 4) {
        // A fragment (16x4 f32, ISA layout): lane l15 row, K-pair by half.
        v2f a; a.x = arow[k0]; a.y = arow[k0 + 1];
        const int kk = k0 + 2 * half;
        // B fragments (4x16 f32): lane -> column n0+l15, VGPRs -> successive K.
        v2f bq; bq.x = qW[kk * D_MODEL + n0 + l15]; bq.y = qW[(kk + 1) * D_MODEL + n0 + l15];
        v2f bk; bk.x = kW[kk * D_MODEL + n0 + l15]; bk.y = kW[(kk + 1) * D_MODEL + n0 + l15];
        v2f bv; bv.x = vW[kk * D_MODEL + n0 + l15]; bv.y = vW[(kk + 1) * D_MODEL + n0 + l15];
        // 8 args: (neg_a, A, neg_b, B, c_mod, C, reuse_a, reuse_b)
        accq = __builtin_amdgcn_wmma_f32_16x16x4_f32(false, a, false, bq, (short)0, accq, false, false);
        acck = __builtin_amdgcn_wmma_f32_16x16x4_f32(false, a, false, bk, (short)0, acck, false, false);
        accv = __builtin_amdgcn_wmma_f32_16x16x4_f32(false, a, false, bv, (short)0, accv, false, false);
    }

    // C/D layout: acc[r] lives at row (r + 8*half), col l15 of the 16x16 tile.
    #pragma unroll
    for (int r = 0; r < 8; ++r) {
        size_t idx = (size_t)(m0 + r + 8 * half) * D_MODEL + n0 + l15;
        Q[idx] = accq[r];
        K[idx] = acck[r];
        V[idx] = accv[r];
    }
}

// ---------------------------------------------------------------------------
// Kernel 2: per-(edge, head) attention score -> exp -> segment-sum into norm.
// ---------------------------------------------------------------------------
__global__ __launch_bounds__(256) void gt_edge_score(
    const float* __restrict__ Q, const float* __restrict__ Kn,
    const long long* __restrict__ rows, const long long* __restrict__ cols,
    float* __restrict__ exp_att, float* __restrict__ norm)
{
    int gid = blockIdx.x * 256 + threadIdx.x;     // gid = e*8 + h
    if (gid >= N_EDGES * N_HEAD) return;
    int e = gid >> 3, h = gid & 7;
    int r = (int)rows[e], c = (int)cols[e];
    const float4* qp = (const float4*)(Q  + (size_t)r * D_MODEL + h * D_HEAD);
    const float4* kp = (const float4*)(Kn + (size_t)c * D_MODEL + h * D_HEAD);
    float att = 0.0f;
    #pragma unroll
    for (int i = 0; i < 4; ++i) {
        float4 a = qp[i], b = kp[i];
        att += a.x * b.x + a.y * b.y + a.z * b.z + a.w * b.w;
    }
    att = fminf(10.0f, fmaxf(-10.0f, att));
    float ex = expf(att);
    exp_att[gid] = ex;
    atomicAdd(&norm[(size_t)r * N_HEAD + h], ex);   // GLOBAL_ATOMIC_ADD_F32
}

// ---------------------------------------------------------------------------
// Kernel 3: normalize and scatter weighted V into res (residual pre-loaded).
// ---------------------------------------------------------------------------
__global__ __launch_bounds__(256) void gt_edge_scatter(
    const float* __restrict__ V,
    const long long* __restrict__ rows, const long long* __restrict__ cols,
    const float* __restrict__ exp_att, const float* __restrict__ norm,
    float* __restrict__ res)
{
    int gid = blockIdx.x * 256 + threadIdx.x;     // gid = e*8 + h
    if (gid >= N_EDGES * N_HEAD) return;
    int e = gid >> 3, h = gid & 7;
    int r = (int)rows[e], c = (int)cols[e];
    float w = exp_att[gid] / (norm[(size_t)r * N_HEAD + h] + 1e-8f);
    const float4* vp = (const float4*)(V + (size_t)c * D_MODEL + h * D_HEAD);
    float* rp = res + (size_t)r * D_MODEL + h * D_HEAD;
    #pragma unroll
    for (int i = 0; i < 4; ++i) {
        float4 v4 = vp[i];
        atomicAdd(rp + 4 * i + 0, w * v4.x);
        atomicAdd(rp + 4 * i + 1, w * v4.y);
        atomicAdd(rp + 4 * i + 2, w * v4.z);
        atomicAdd(rp + 4 * i + 3, w * v4.w);
    }
}

// ---------------------------------------------------------------------------
// Kernel 4: LayerNorm, one wave32 per node (4 floats/lane, shfl_xor reduce).
// ---------------------------------------------------------------------------
__global__ __launch_bounds__(256) void gt_layernorm(
    const float* __restrict__ res, const float* __restrict__ gamma,
    const float* __restrict__ beta, float* __restrict__ out)
{
    int node = blockIdx.x * 8 + (threadIdx.x >> 5);   // 12500 * 8 = 100000 exactly
    int lane = threadIdx.x & 31;
    float4 x = ((const float4*)(res + (size_t)node * D_MODEL))[lane];
    float s  = x.x + x.y + x.z + x.w;
    float ss = x.x * x.x + x.y * x.y + x.z * x.z + x.w * x.w;
    #pragma unroll
    for (int m = 16; m >= 1; m >>= 1) {
        s  += __shfl_xor(s,  m, 32);
        ss += __shfl_xor(ss, m, 32);
    }
    float mean = s * (1.0f / 128.0f);
    float var  = ss * (1.0f / 128.0f) - mean * mean;
    float inv  = rsqrtf(var + LN_EPS);
    float4 g = ((const float4*)gamma)[lane];
    float4 b = ((const float4*)beta)[lane];
    float4 o;
    o.x = (x.x - mean) * inv * g.x + b.x;
    o.y = (x.y - mean) * inv * g.y + b.y;
    o.z = (x.z - mean) * inv * g.z + b.z;
    o.w = (x.w - mean) * inv * g.w + b.w;
    ((float4*)(out + (size_t)node * D_MODEL))[lane] = o;
}

// ---------------------------------------------------------------------------
extern "C" void kernel_launch(void* const* d_in, const int* in_sizes, int n_in,
                              void* d_out, int out_size, void* d_ws, size_t ws_size,
                              hipStream_t stream)
{
    const float*     embeds = (const float*)d_in[0];
    const float*     qW     = (const float*)d_in[1];
    const float*     kW     = (const float*)d_in[2];
    const float*     vW     = (const float*)d_in[3];
    const float*     gamma  = (const float*)d_in[4];
    const float*     beta   = (const float*)d_in[5];
    const long long* eidx   = (const long long*)d_in[6];   // int64 [2, E]
    const long long* rows   = eidx;
    const long long* cols   = eidx + N_EDGES;

    // Workspace layout (floats): Q | K | V | res | norm | exp_att
    const size_t NR = (size_t)N_NODES * D_MODEL;           // 12,800,000
    float* ws      = (float*)d_ws;
    float* Q       = ws;
    float* Kbuf    = ws + NR;
    float* Vbuf    = ws + 2 * NR;
    float* res     = ws + 3 * NR;
    float* norm    = ws + 4 * NR;                          // N_NODES * N_HEAD
    float* exp_att = ws + 4 * NR + (size_t)N_NODES * N_HEAD;

    // Phase 1: node-level fused QKV GEMM (WMMA f32), res/norm init.
    gt_qkv_gemm<<<N_NODES / 16, 256, 0, stream>>>(
        embeds, qW, kW, vW, Q, Kbuf, Vbuf, res, norm);

    // Phase 2: edge scores + segment-sum of exp.
    const int eh = N_EDGES * N_HEAD;                       // 6,400,000
    gt_edge_score<<<(eh + 255) / 256, 256, 0, stream>>>(
        Q, Kbuf, rows, cols, exp_att, norm);

    // Phase 3: normalized weighted scatter of V.
    gt_edge_scatter<<<(eh + 255) / 256, 256, 0, stream>>>(
        Vbuf, rows, cols, exp_att, norm, res);

    // Phase 4: LayerNorm into d_out.
    gt_layernorm<<<N_NODES / 8, 256, 0, stream>>>(
        res, gamma, beta, (float*)d_out);
}